// GRNByIrrep_71322226918038
// MI455X (gfx1250) — compile-verified
//
#include <hip/hip_runtime.h>
#include <cstdint>
#include <cstddef>

// GRN-by-irrep: (16, 480, 64, 64) fp32.
// FIELD_SIZES = tile([1,1,2,2,4,4,8,8], 16) -> 128 fields, 30 channels/tile, 480 channels.
// Memory-bound (~0.4 FLOP/byte): optimize pure data movement.
//   pass1: sumsq per (b,c) plane via async global->LDS b128 copies (RT hint -> x lands in L2;
//          126 MB fits in the 192 MB L2, so pass3's re-read of x hits L2).
//   pass2: tiny stats kernel -> per-channel scale = gamma*nx + 1.
//   pass3: out = x*scale + beta, 128-bit loads, non-temporal 128-bit stores (don't evict x).

#define B_  16
#define C_  480
#define F_  128
#define HW_ 4096   // 64*64 floats per plane
#define NPLANE (B_ * C_)   // 7680

typedef float v4f __attribute__((ext_vector_type(4)));

__device__ __forceinline__ uint32_t lds_off_u32(const void* p) {
  // Generic LDS pointer = {LDS aperture | lds_offset[31:0]} -> low 32 bits are the DS address.
  return (uint32_t)(uintptr_t)p;
}

// ---------------------------------------------------------------------------
// Pass 1: per-plane sum of squares. One block per (b,c) plane (16 KB).
// Each wave issues 4 async b128 copies (one 16B slice per lane per chunk),
// drains them in order with s_wait_asynccnt, reads back its own bytes from LDS.
// ---------------------------------------------------------------------------
__global__ __launch_bounds__(256) void grn_sumsq(const float* __restrict__ x,
                                                 float* __restrict__ sq) {
  __shared__ float tile[HW_];        // 16 KB: whole plane, 4 chunks of 4 KB
  const int bid = blockIdx.x;        // plane id = b*C_ + c
  const int tid = threadIdx.x;       // 0..255

  const uint64_t gaddr = (uint64_t)(uintptr_t)(x + (size_t)bid * HW_) + (uint32_t)(tid * 16);
  const uint32_t laddr = lds_off_u32(tile) + (uint32_t)(tid * 16);

  // Issue all 4 chunk copies up front (ASYNCcnt = 4 per wave). offset: is added
  // to both the LDS destination and the global source (ISA 08 §4.4).
  asm volatile("global_load_async_to_lds_b128 %0, %1, off"              :: "v"(laddr), "v"(gaddr) : "memory");
  asm volatile("global_load_async_to_lds_b128 %0, %1, off offset:4096"  :: "v"(laddr), "v"(gaddr) : "memory");
  asm volatile("global_load_async_to_lds_b128 %0, %1, off offset:8192"  :: "v"(laddr), "v"(gaddr) : "memory");
  asm volatile("global_load_async_to_lds_b128 %0, %1, off offset:12288" :: "v"(laddr), "v"(gaddr) : "memory");

  float acc = 0.0f;
  // Async loads complete in order within a wave -> drain chunk by chunk.
  asm volatile("s_wait_asynccnt 3" ::: "memory");
  {
    v4f v = *(const v4f*)&tile[0 * 1024 + tid * 4];
    acc += v.x * v.x + v.y * v.y + v.z * v.z + v.w * v.w;
  }
  asm volatile("s_wait_asynccnt 2" ::: "memory");
  {
    v4f v = *(const v4f*)&tile[1 * 1024 + tid * 4];
    acc += v.x * v.x + v.y * v.y + v.z * v.z + v.w * v.w;
  }
  asm volatile("s_wait_asynccnt 1" ::: "memory");
  {
    v4f v = *(const v4f*)&tile[2 * 1024 + tid * 4];
    acc += v.x * v.x + v.y * v.y + v.z * v.z + v.w * v.w;
  }
  asm volatile("s_wait_asynccnt 0" ::: "memory");
  {
    v4f v = *(const v4f*)&tile[3 * 1024 + tid * 4];
    acc += v.x * v.x + v.y * v.y + v.z * v.z + v.w * v.w;
  }

  // Wave32 shuffle reduction, then cross-wave via LDS.
  #pragma unroll
  for (int off = 16; off > 0; off >>= 1) acc += __shfl_down(acc, off, 32);

  __shared__ float wred[8];
  if ((tid & 31) == 0) wred[tid >> 5] = acc;
  __syncthreads();
  if (tid == 0) {
    float s = 0.0f;
    #pragma unroll
    for (int i = 0; i < 8; ++i) s += wred[i];
    sq[bid] = s;
  }
}

// ---------------------------------------------------------------------------
// Pass 2: per (batch, field): gx = sqrt(segment_sum(sq)); nx = gx/(mean_f(gx)+eps);
// write per-channel scale = gamma_f * nx + 1. One block per batch, thread==field.
// ---------------------------------------------------------------------------
__global__ __launch_bounds__(F_) void grn_stats(const float* __restrict__ sq,
                                                const float* __restrict__ gamma,
                                                float* __restrict__ scaleA) {
  const int b = blockIdx.x;    // 0..15
  const int f = threadIdx.x;   // 0..127
  const int k = f & 7;         // field slot within irrep tile
  const int t = f >> 3;        // irrep tile 0..15

  const int offs[8] = {0, 1, 2, 4, 6, 10, 14, 22};
  const int szs[8]  = {1, 1, 2, 2, 4, 4, 8, 8};
  const int c0 = t * 30 + offs[k];
  const int n  = szs[k];

  float s = 0.0f;
  for (int i = 0; i < n; ++i) s += sq[b * C_ + c0 + i];
  const float gx = sqrtf(s);

  __shared__ float red[F_];
  red[f] = gx;
  __syncthreads();
  #pragma unroll
  for (int st = F_ / 2; st > 0; st >>= 1) {
    if (f < st) red[f] += red[f + st];
    __syncthreads();
  }
  const float mean = red[0] * (1.0f / (float)F_);
  const float nx = gx / (mean + 1e-6f);
  const float scale = gamma[f] * nx + 1.0f;   // out = x*scale + beta

  for (int i = 0; i < n; ++i) scaleA[b * C_ + c0 + i] = scale;
}

// ---------------------------------------------------------------------------
// Pass 3: out = x*scale + beta. One block per (b,c) plane; b128 loads (hit L2),
// non-temporal b128 stores (keep x resident in L2).
// ---------------------------------------------------------------------------
__global__ __launch_bounds__(256) void grn_apply(const float* __restrict__ x,
                                                 const float* __restrict__ scaleA,
                                                 const float* __restrict__ beta,
                                                 float* __restrict__ out) {
  const int bid = blockIdx.x;        // plane id = b*C_ + c
  const int tid = threadIdx.x;
  const int c = bid % C_;

  // channel -> field: tile of 30 channels, boundaries 0,1,2,4,6,10,14,22.
  const int t = c / 30;
  const int r = c - t * 30;
  const int k = (r < 1) ? 0 : (r < 2) ? 1 : (r < 4) ? 2 : (r < 6) ? 3
              : (r < 10) ? 4 : (r < 14) ? 5 : (r < 22) ? 6 : 7;

  const float sc = scaleA[bid];
  const float bi = beta[t * 8 + k];

  const v4f* __restrict__ xin = (const v4f*)(x + (size_t)bid * HW_);
  v4f* __restrict__ o = (v4f*)(out + (size_t)bid * HW_);

  #pragma unroll
  for (int i = 0; i < 4; ++i) {
    const int idx = tid + 256 * i;
    v4f v = xin[idx];
    v4f w = v * sc + bi;
    __builtin_nontemporal_store(w, &o[idx]);
  }
}

// ---------------------------------------------------------------------------
extern "C" void kernel_launch(void* const* d_in, const int* in_sizes, int n_in,
                              void* d_out, int out_size, void* d_ws, size_t ws_size,
                              hipStream_t stream) {
  (void)in_sizes; (void)n_in; (void)out_size; (void)ws_size;
  const float* x     = (const float*)d_in[0];   // (16,480,64,64)
  const float* gamma = (const float*)d_in[1];   // 128
  const float* beta  = (const float*)d_in[2];   // 128
  float* out = (float*)d_out;

  float* sq     = (float*)d_ws;        // NPLANE floats
  float* scaleA = sq + NPLANE;         // NPLANE floats

  grn_sumsq<<<NPLANE, 256, 0, stream>>>(x, sq);
  grn_stats<<<B_, F_, 0, stream>>>(sq, gamma, scaleA);
  grn_apply<<<NPLANE, 256, 0, stream>>>(x, scaleA, beta, out);
}